// GNN_87935160418912
// MI455X (gfx1250) — compile-verified
//
#include <hip/hip_runtime.h>
#include <hip/hip_bf16.h>
#include <math.h>

#define DIM 128
#define NLAYERS 3

typedef __attribute__((ext_vector_type(2))) float v2f;
typedef __attribute__((ext_vector_type(8))) float v8f;

// ---------------------------------------------------------------------------
// agg = h  (float4 copy)
// ---------------------------------------------------------------------------
__global__ void gnn_copy_kernel(const float4* __restrict__ src4,
                                float4* __restrict__ dst4, int n4) {
    int i = blockIdx.x * blockDim.x + threadIdx.x;
    if (i < n4) dst4[i] = src4[i];
}

// ---------------------------------------------------------------------------
// agg[dst[e]] += h[src[e]]   (32 lanes per edge, float4 chunk per lane)
// ---------------------------------------------------------------------------
__global__ void gnn_scatter_kernel(const float* __restrict__ h,
                                   const int* __restrict__ src,
                                   const int* __restrict__ dst,
                                   float* __restrict__ agg, int n_edges) {
    int t = blockIdx.x * blockDim.x + threadIdx.x;
    int e = t >> 5;
    int q = t & 31;                       // 32 * float4 = 128 floats
    if (e >= n_edges) return;
    int s = src[e];
    int d = dst[e];
    float4 v = ((const float4*)(h + (size_t)s * DIM))[q];
    float* o = agg + (size_t)d * DIM + q * 4;
    unsafeAtomicAdd(o + 0, v.x);
    unsafeAtomicAdd(o + 1, v.y);
    unsafeAtomicAdd(o + 2, v.z);
    unsafeAtomicAdd(o + 3, v.w);
}

// ---------------------------------------------------------------------------
// zero a small vector
// ---------------------------------------------------------------------------
__global__ void gnn_zero_kernel(float* __restrict__ p, int n) {
    int i = blockIdx.x * blockDim.x + threadIdx.x;
    if (i < n) p[i] = 0.0f;
}

// ---------------------------------------------------------------------------
// ro[c] = sum_n h[n][c]   (per-block partial sums, 128 atomics per block)
// ---------------------------------------------------------------------------
#define COLSUM_ROWS 250
__global__ void gnn_colsum_kernel(const float* __restrict__ h,
                                  float* __restrict__ ro, int n_nodes) {
    int col = threadIdx.x;                // 0..127
    int r0 = blockIdx.x * COLSUM_ROWS;
    int r1 = r0 + COLSUM_ROWS;
    if (r1 > n_nodes) r1 = n_nodes;
    float s = 0.0f;
    for (int r = r0; r < r1; ++r) s += h[(size_t)r * DIM + col];
    unsafeAtomicAdd(&ro[col], s);
}

// ---------------------------------------------------------------------------
// cvec[j] = Vb[j] + Ab[j] + Rb[j] + dot(ro, Rw[j,:])   (one block, 128 thr)
// ---------------------------------------------------------------------------
__global__ void gnn_cvec_kernel(const float* __restrict__ ro,
                                const float* __restrict__ Rw,
                                const float* __restrict__ Vb,
                                const float* __restrict__ Ab,
                                const float* __restrict__ Rb,
                                float* __restrict__ cvec) {
    int j = threadIdx.x;                  // 0..127
    float s = Vb[j] + Ab[j] + Rb[j];
    const float* rw = Rw + (size_t)j * DIM;
    for (int k = 0; k < DIM; k += 4) {
        s += ro[k] * rw[k] + ro[k + 1] * rw[k + 1] +
             ro[k + 2] * rw[k + 2] + ro[k + 3] * rw[k + 3];
    }
    cvec[j] = s;
}

// ---------------------------------------------------------------------------
// hnext = relu(h @ Vw^T + agg @ Aw^T + cvec)  via V_WMMA_F32_16X16X4_F32.
// Block = 256 threads = 8 waves; each wave owns a 16-node row tile and keeps
// all 8 column-tile accumulators (64 VGPRs) live.  k is the OUTER loop so
// each A fragment (h row pair, agg row pair) is loaded once and feeds 8
// WMMAs; B fragments stream from LDS-staged weights (128 KB dynamic LDS).
// ---------------------------------------------------------------------------
__global__ void __launch_bounds__(256)
gnn_combine_wmma_kernel(const float* __restrict__ h,
                        const float* __restrict__ agg,
                        const float* __restrict__ Vw,
                        const float* __restrict__ Aw,
                        const float* __restrict__ cvec,
                        float* __restrict__ hnext,
                        int n_nodes) {
    extern __shared__ float lds[];
    float* ldsV = lds;                    // Vw[j][k], row-major 128x128
    float* ldsA = lds + DIM * DIM;

    // cooperative weight staging (float4)
    {
        const float4* v4 = (const float4*)Vw;
        const float4* a4 = (const float4*)Aw;
        float4* lv4 = (float4*)ldsV;
        float4* la4 = (float4*)ldsA;
        for (int i = threadIdx.x; i < DIM * DIM / 4; i += blockDim.x) {
            lv4[i] = v4[i];
            la4[i] = a4[i];
        }
    }
    __syncthreads();

    const int wave = threadIdx.x >> 5;    // 0..7
    const int lane = threadIdx.x & 31;
    const int row_base = (blockIdx.x * 8 + wave) * 16;
    if (row_base >= n_nodes) return;      // wave-uniform: EXEC stays all-ones

    const int lr = lane & 15;             // 0..15
    const int hi = lane >> 4;             // 0 or 1

    // A-matrix rows for this lane (16x4 fp32 layout: lanes 0-15 K={k,k+1},
    // lanes 16-31 K={k+2,k+3})
    const float* hrow = h   + (size_t)(row_base + lr) * DIM;
    const float* grow = agg + (size_t)(row_base + lr) * DIM;
    // B = W^T: element (K=k, N=j) = W[j][k]; lane's column is j0 + lr
    const float* vcol = ldsV + (size_t)lr * DIM;
    const float* acol = ldsA + (size_t)lr * DIM;

    v8f acc[8];
#pragma unroll
    for (int j = 0; j < 8; ++j) acc[j] = (v8f){};

#pragma unroll 4
    for (int k0 = 0; k0 < DIM; k0 += 4) {
        const int k = k0 + 2 * hi;
        v2f a_h = {hrow[k], hrow[k + 1]};   // one global b64 each, reused 8x
        v2f a_g = {grow[k], grow[k + 1]};
#pragma unroll
        for (int j = 0; j < 8; ++j) {
            const float* vc = vcol + (size_t)j * 16 * DIM;
            const float* ac = acol + (size_t)j * 16 * DIM;
            v2f b_v = {vc[k], vc[k + 1]};
            v2f b_a = {ac[k], ac[k + 1]};
            acc[j] = __builtin_amdgcn_wmma_f32_16x16x4_f32(
                false, a_h, false, b_v, (short)0, acc[j], false, false);
            acc[j] = __builtin_amdgcn_wmma_f32_16x16x4_f32(
                false, a_g, false, b_a, (short)0, acc[j], false, false);
        }
    }

    // epilogue: + broadcast term, relu, store
    // C/D layout: VGPR i -> row (row_base + i + 8*hi), col = j*16 + lr
#pragma unroll
    for (int j = 0; j < 8; ++j) {
        const float c = cvec[j * 16 + lr];
        float* outp = hnext + (size_t)(row_base + 8 * hi) * DIM + j * 16 + lr;
#pragma unroll
        for (int i = 0; i < 8; ++i) {
            float v = acc[j][i] + c;
            v = v > 0.0f ? v : 0.0f;
            outp[(size_t)i * DIM] = v;
        }
    }
}

// ---------------------------------------------------------------------------
// out[n][j] = sigmoid(h[n] . out_w[j] + out_b[j]),  j in {0,1}
// ---------------------------------------------------------------------------
__global__ void gnn_head_kernel(const float* __restrict__ h,
                                const float* __restrict__ ow,
                                const float* __restrict__ ob,
                                float* __restrict__ out, int n_nodes) {
    int i = blockIdx.x * blockDim.x + threadIdx.x;
    if (i >= n_nodes) return;
    const float4* hr = (const float4*)(h + (size_t)i * DIM);
    const float4* w0 = (const float4*)ow;
    const float4* w1 = (const float4*)(ow + DIM);
    float s0 = 0.0f, s1 = 0.0f;
    for (int q = 0; q < DIM / 4; ++q) {
        float4 v = hr[q], a = w0[q], b = w1[q];
        s0 += v.x * a.x + v.y * a.y + v.z * a.z + v.w * a.w;
        s1 += v.x * b.x + v.y * b.y + v.z * b.z + v.w * b.w;
    }
    s0 += ob[0];
    s1 += ob[1];
    out[(size_t)i * 2 + 0] = 1.0f / (1.0f + __expf(-s0));
    out[(size_t)i * 2 + 1] = 1.0f / (1.0f + __expf(-s1));
}

// ---------------------------------------------------------------------------
// host-side orchestration
// ---------------------------------------------------------------------------
extern "C" void kernel_launch(void* const* d_in, const int* in_sizes, int n_in,
                              void* d_out, int out_size, void* d_ws, size_t ws_size,
                              hipStream_t stream) {
    const float* x   = (const float*)d_in[0];
    const int*   src = (const int*)d_in[1];
    const int*   dst = (const int*)d_in[2];
    const float* Vw  = (const float*)d_in[3];
    const float* Vb  = (const float*)d_in[4];
    const float* Aw  = (const float*)d_in[5];
    const float* Ab  = (const float*)d_in[6];
    const float* Rw  = (const float*)d_in[7];
    const float* Rb  = (const float*)d_in[8];
    const float* ow  = (const float*)d_in[9];
    const float* ob  = (const float*)d_in[10];
    float* out = (float*)d_out;

    const int N = in_sizes[0] / DIM;   // 50000
    const int E = in_sizes[1];         // 800000
    const size_t feat = (size_t)N * DIM;

    float* buf0 = (float*)d_ws;
    float* buf1 = buf0 + feat;
    float* agg  = buf1 + feat;
    float* ro   = agg + feat;
    float* cvec = ro + DIM;

    const int n4 = (int)(feat / 4);
    const int copy_blocks    = (n4 + 255) / 256;
    const int scatter_blocks = (int)(((size_t)E * 32 + 255) / 256);
    const int colsum_blocks  = (N + COLSUM_ROWS - 1) / COLSUM_ROWS;
    const int row_tiles      = (N + 15) / 16;
    const int combine_blocks = (row_tiles + 7) / 8;
    const size_t lds_bytes   = 2 * DIM * DIM * sizeof(float);   // 128 KB

    const float* hcur = x;
    for (int l = 0; l < NLAYERS; ++l) {
        float* hnext = (l & 1) ? buf1 : buf0;
        const float* Vw_l = Vw + (size_t)l * DIM * DIM;
        const float* Aw_l = Aw + (size_t)l * DIM * DIM;
        const float* Rw_l = Rw + (size_t)l * DIM * DIM;
        const float* Vb_l = Vb + (size_t)l * DIM;
        const float* Ab_l = Ab + (size_t)l * DIM;
        const float* Rb_l = Rb + (size_t)l * DIM;

        // agg = hcur (self loop), then scatter-add neighbors
        gnn_copy_kernel<<<copy_blocks, 256, 0, stream>>>(
            (const float4*)hcur, (float4*)agg, n4);
        gnn_scatter_kernel<<<scatter_blocks, 256, 0, stream>>>(
            hcur, src, dst, agg, E);

        // readout
        gnn_zero_kernel<<<1, DIM, 0, stream>>>(ro, DIM);
        gnn_colsum_kernel<<<colsum_blocks, DIM, 0, stream>>>(hcur, ro, N);
        gnn_cvec_kernel<<<1, DIM, 0, stream>>>(ro, Rw_l, Vb_l, Ab_l, Rb_l, cvec);

        // fused double-GEMM + bias + relu via fp32 WMMA
        gnn_combine_wmma_kernel<<<combine_blocks, 256, lds_bytes, stream>>>(
            hcur, agg, Vw_l, Aw_l, cvec, hnext, N);

        hcur = hnext;
    }

    gnn_head_kernel<<<(N + 255) / 256, 256, 0, stream>>>(hcur, ow, ob, out, N);
}